// GAT_23149873725487
// MI455X (gfx1250) — compile-verified
//
#include <hip/hip_runtime.h>
#include <math.h>

#define N_NODES 50000
#define N_EDGES 800000
#define F_INC   256
#define HIDC    96
#define NCLSC   32
#define NEG_SLOPE 0.2f

typedef float v2f __attribute__((ext_vector_type(2)));
typedef float v8f __attribute__((ext_vector_type(8)));

// ---------------- wave32 reductions ----------------
__device__ __forceinline__ float wave_sum(float v) {
  #pragma unroll
  for (int off = 16; off > 0; off >>= 1) v += __shfl_xor(v, off, 32);
  return v;
}
__device__ __forceinline__ float wave_max(float v) {
  #pragma unroll
  for (int off = 16; off > 0; off >>= 1) v = fmaxf(v, __shfl_xor(v, off, 32));
  return v;
}

// float atomic max via int/uint ordering trick (dest pre-initialized to -inf)
__device__ __forceinline__ void atomic_max_float(float* addr, float val) {
  if (val >= 0.0f) atomicMax((int*)addr, __float_as_int(val));
  else             atomicMin((unsigned int*)addr, __float_as_uint(val));
}

// ---------------- utility fills ----------------
__global__ void fill_kernel(float* __restrict__ p, float v, int n) {
  for (int i = blockIdx.x * blockDim.x + threadIdx.x; i < n; i += gridDim.x * blockDim.x)
    p[i] = v;
}

// ---------------- fp32 WMMA GEMM: Out[N x NCOL] = A[N x K] @ W[K x NCOL] ----------------
// 5 waves/block, one 16-row stripe per wave, all NCOL columns per wave.
// Grid must be exactly (nStripes/5) so every wave has full EXEC at the WMMAs.
template<int K, int NCOL, int KC>
__global__ void __launch_bounds__(160)
gemm_wmma_kernel(const float* __restrict__ A, const float* __restrict__ W,
                 float* __restrict__ Out) {
  __shared__ float ldsW[KC * NCOL];

  const int lane  = threadIdx.x & 31;
  const int wave  = threadIdx.x >> 5;          // 0..4
  const int half  = lane >> 4;                 // 0 or 1
  const int mlan  = lane & 15;                 // row-in-tile / col-in-tile
  const int stripe = blockIdx.x * 5 + wave;    // exact coverage, no tail
  const int r0 = stripe * 16;

  constexpr int NT = NCOL / 16;
  v8f acc[NT];
  #pragma unroll
  for (int t = 0; t < NT; ++t) acc[t] = v8f{0.f,0.f,0.f,0.f,0.f,0.f,0.f,0.f};

  const float* arow = A + (size_t)(r0 + mlan) * K;

  for (int kc = 0; kc < K; kc += KC) {
    __syncthreads();
    for (int i = threadIdx.x; i < KC * NCOL; i += 160)
      ldsW[i] = W[kc * NCOL + i];
    __syncthreads();

    for (int k0 = 0; k0 < KC; k0 += 4) {
      // A 16x4 fp32 tile: VGPR0 = K={0,2}+base, VGPR1 = K={1,3}+base (halves by lane>>4)
      const float* ap = arow + kc + k0 + 2 * half;
      v2f a; a.x = ap[0]; a.y = ap[1];
      const float* w0 = &ldsW[(k0 + 2 * half) * NCOL];
      const float* w1 = w0 + NCOL;
      #pragma unroll
      for (int t = 0; t < NT; ++t) {
        v2f b;
        b.x = w0[t * 16 + mlan];
        b.y = w1[t * 16 + mlan];
        acc[t] = __builtin_amdgcn_wmma_f32_16x16x4_f32(
            false, a, false, b, (short)0, acc[t], false, false);
      }
    }
  }

  // C/D layout: VGPR r -> M = r + 8*half, N = mlan
  #pragma unroll
  for (int t = 0; t < NT; ++t)
    #pragma unroll
    for (int r = 0; r < 8; ++r)
      Out[(size_t)(r0 + r + 8 * half) * NCOL + t * 16 + mlan] = acc[t][r];
}

// ---------------- per-node alpha_src / alpha_dst (wave per node) ----------------
template<int F>
__global__ void alpha_kernel(const float* __restrict__ h,
                             const float* __restrict__ a_src,
                             const float* __restrict__ a_dst,
                             float* __restrict__ out_s, float* __restrict__ out_d) {
  const int wid  = (blockIdx.x * blockDim.x + threadIdx.x) >> 5;
  const int lane = threadIdx.x & 31;
  if (wid >= N_NODES) return;
  float ss = 0.f, sd = 0.f;
  #pragma unroll
  for (int f = lane; f < F; f += 32) {
    float hv = h[(size_t)wid * F + f];
    ss += hv * a_src[f];
    sd += hv * a_dst[f];
  }
  ss = wave_sum(ss);
  sd = wave_sum(sd);
  if (lane == 0) { out_s[wid] = ss; out_d[wid] = sd; }
}

// ---------------- edge pass 1: segment max of leaky-relu logits ----------------
__global__ void edge_max_kernel(const long long* __restrict__ ei,
                                const float* __restrict__ as, const float* __restrict__ ad,
                                float* __restrict__ nmax) {
  const int etot = N_EDGES + N_NODES;
  for (int e = blockIdx.x * blockDim.x + threadIdx.x; e < etot; e += gridDim.x * blockDim.x) {
    int s, d;
    if (e < N_EDGES) { s = (int)ei[e]; d = (int)ei[N_EDGES + e]; }
    else             { s = d = e - N_EDGES; }
    float v = as[s] + ad[d];
    v = v > 0.f ? v : NEG_SLOPE * v;
    atomic_max_float(&nmax[d], v);
  }
}

// ---------------- edge pass 2: exp(e - max[dst]) and segment sum ----------------
__global__ void edge_expsum_kernel(const long long* __restrict__ ei,
                                   const float* __restrict__ as, const float* __restrict__ ad,
                                   const float* __restrict__ nmax,
                                   float* __restrict__ exbuf, float* __restrict__ nsum) {
  const int etot = N_EDGES + N_NODES;
  for (int e = blockIdx.x * blockDim.x + threadIdx.x; e < etot; e += gridDim.x * blockDim.x) {
    int s, d;
    if (e < N_EDGES) { s = (int)ei[e]; d = (int)ei[N_EDGES + e]; }
    else             { s = d = e - N_EDGES; }
    float v = as[s] + ad[d];
    v = v > 0.f ? v : NEG_SLOPE * v;
    float ex = __expf(v - nmax[d]);
    exbuf[e] = ex;
    atomicAdd(&nsum[d], ex);
  }
}

// ---------------- edge pass 3: weighted aggregate (wave per edge) ----------------
template<int F>
__global__ void edge_agg_kernel(const long long* __restrict__ ei,
                                const float* __restrict__ exbuf, const float* __restrict__ nsum,
                                const float* __restrict__ h, float* __restrict__ agg) {
  const int etot = N_EDGES + N_NODES;
  const int lane = threadIdx.x & 31;
  const int wid  = (blockIdx.x * blockDim.x + threadIdx.x) >> 5;
  const int nw   = (gridDim.x * blockDim.x) >> 5;
  for (int e = wid; e < etot; e += nw) {
    int s, d;
    if (e < N_EDGES) { s = (int)ei[e]; d = (int)ei[N_EDGES + e]; }
    else             { s = d = e - N_EDGES; }
    float alpha = exbuf[e] / nsum[d];
    #pragma unroll
    for (int f = lane; f < F; f += 32)
      atomicAdd(&agg[(size_t)d * F + f], alpha * h[(size_t)s * F + f]);
  }
}

// ---------------- elementwise: x = elu(x + b) (in place) ----------------
__global__ void elu_bias_kernel(float* __restrict__ x, const float* __restrict__ b) {
  const int n = N_NODES * HIDC;
  for (int i = blockIdx.x * blockDim.x + threadIdx.x; i < n; i += gridDim.x * blockDim.x) {
    float v = x[i] + b[i % HIDC];
    x[i] = v > 0.f ? v : (__expf(v) - 1.f);
  }
}

// ---------------- final: out = log_softmax(agg + b2) (wave per node, lane = class) ----------------
__global__ void logsoftmax_kernel(const float* __restrict__ agg, const float* __restrict__ b2,
                                  float* __restrict__ out) {
  const int wid  = (blockIdx.x * blockDim.x + threadIdx.x) >> 5;
  const int lane = threadIdx.x & 31;
  if (wid >= N_NODES) return;
  float v = agg[(size_t)wid * NCLSC + lane] + b2[lane];
  float m = wave_max(v);
  float ex = __expf(v - m);
  float s = wave_sum(ex);
  out[(size_t)wid * NCLSC + lane] = v - m - __logf(s);
}

extern "C" void kernel_launch(void* const* d_in, const int* in_sizes, int n_in,
                              void* d_out, int out_size, void* d_ws, size_t ws_size,
                              hipStream_t stream) {
  const float*     x   = (const float*)d_in[0];
  const long long* ei  = (const long long*)d_in[1];   // int64 [2, E]
  const float*     W1  = (const float*)d_in[2];
  const float*     a1s = (const float*)d_in[3];
  const float*     a1d = (const float*)d_in[4];
  const float*     b1  = (const float*)d_in[5];
  const float*     W2  = (const float*)d_in[6];
  const float*     a2s = (const float*)d_in[7];
  const float*     a2d = (const float*)d_in[8];
  const float*     b2  = (const float*)d_in[9];
  float* out = (float*)d_out;

  // workspace carve-up (~55.4 MB of floats)
  float* h1    = (float*)d_ws;                         // N*96
  float* agg1  = h1   + (size_t)N_NODES * HIDC;        // N*96 (becomes ELU'd layer-2 input)
  float* h2    = agg1 + (size_t)N_NODES * HIDC;        // N*32
  float* agg2  = h2   + (size_t)N_NODES * NCLSC;       // N*32
  float* asb   = agg2 + (size_t)N_NODES * NCLSC;       // N
  float* adb   = asb  + N_NODES;                       // N
  float* nmax  = adb  + N_NODES;                       // N
  float* nsum  = nmax + N_NODES;                       // N
  float* exbuf = nsum + N_NODES;                       // E + N

  const float NEG_INF = -__builtin_huge_valf();

  // ----- layer 1 -----
  fill_kernel<<<1024, 256, 0, stream>>>(agg1, 0.f, N_NODES * HIDC);
  fill_kernel<<<256, 256, 0, stream>>>(nmax, NEG_INF, N_NODES);
  fill_kernel<<<256, 256, 0, stream>>>(nsum, 0.f, N_NODES);

  gemm_wmma_kernel<F_INC, HIDC, 64><<<625, 160, 0, stream>>>(x, W1, h1);
  alpha_kernel<HIDC><<<6250, 256, 0, stream>>>(h1, a1s, a1d, asb, adb);

  edge_max_kernel<<<2048, 256, 0, stream>>>(ei, asb, adb, nmax);
  edge_expsum_kernel<<<2048, 256, 0, stream>>>(ei, asb, adb, nmax, exbuf, nsum);
  edge_agg_kernel<HIDC><<<4096, 256, 0, stream>>>(ei, exbuf, nsum, h1, agg1);

  elu_bias_kernel<<<2048, 256, 0, stream>>>(agg1, b1);

  // ----- layer 2 -----
  fill_kernel<<<512, 256, 0, stream>>>(agg2, 0.f, N_NODES * NCLSC);
  fill_kernel<<<256, 256, 0, stream>>>(nmax, NEG_INF, N_NODES);
  fill_kernel<<<256, 256, 0, stream>>>(nsum, 0.f, N_NODES);

  gemm_wmma_kernel<HIDC, NCLSC, HIDC><<<625, 160, 0, stream>>>(agg1, W2, h2);
  alpha_kernel<NCLSC><<<6250, 256, 0, stream>>>(h2, a2s, a2d, asb, adb);

  edge_max_kernel<<<2048, 256, 0, stream>>>(ei, asb, adb, nmax);
  edge_expsum_kernel<<<2048, 256, 0, stream>>>(ei, asb, adb, nmax, exbuf, nsum);
  edge_agg_kernel<NCLSC><<<4096, 256, 0, stream>>>(ei, exbuf, nsum, h2, agg2);

  logsoftmax_kernel<<<6250, 256, 0, stream>>>(agg2, b2, out);
}